// CrossAttentionLayer_28217935134789
// MI455X (gfx1250) — compile-verified
//
#include <hip/hip_runtime.h>
#include <hip/hip_bf16.h>
#include <math.h>

typedef __attribute__((ext_vector_type(2))) float v2f;
typedef __attribute__((ext_vector_type(4))) float v4f;
typedef __attribute__((ext_vector_type(8))) float v8f;

#define HIDDEN 2048
#define D_MODEL 1024
#define N_HEADS 16
#define D_K 64
#define SEQ 2048
#define BATCH 32
#define INV_SCALE (1.0f / 12.0f)   /* 1/(sqrt(64)*1.5) */
#define LN_EPS 1e-5f

/* ---------------- kernel 1: Q[b,i] = query[b,:] . W_q[i,:]  (i in [0,1024)) */
__global__ void qproj_kernel(const float* __restrict__ query,
                             const float* __restrict__ Wq,
                             float* __restrict__ Q) {
  int b = blockIdx.y;
  int i = blockIdx.x * blockDim.x + threadIdx.x;
  const float* q = query + (size_t)b * HIDDEN;
  const float* w = Wq + (size_t)i * HIDDEN;
  float acc = 0.f;
  for (int k = 0; k < HIDDEN; k += 4) {
    v4f wv = *(const v4f*)(w + k);
    v4f qv = *(const v4f*)(q + k);
    acc += wv.x * qv.x + wv.y * qv.y + wv.z * qv.z + wv.w * qv.w;
  }
  Q[(size_t)b * D_MODEL + i] = acc;
}

/* ---------------- kernel 2: U[b,h,k] = sum_d Q[b,h,d] * W_k[h*64+d, k] */
__global__ void ukey_kernel(const float* __restrict__ Q,
                            const float* __restrict__ Wk,
                            float* __restrict__ U) {
  int b = blockIdx.y;
  int k = blockIdx.x * blockDim.x + threadIdx.x; /* 0..2047, coalesced */
  const float* qb = Q + (size_t)b * D_MODEL;
  for (int h = 0; h < N_HEADS; ++h) {
    float acc = 0.f;
    const float* w = Wk + (size_t)h * D_K * HIDDEN + k;
    #pragma unroll 8
    for (int d = 0; d < D_K; ++d)
      acc += qb[h * D_K + d] * w[(size_t)d * HIDDEN];
    U[((size_t)b * N_HEADS + h) * HIDDEN + k] = acc;
  }
}

/* ---------------- kernel 3 (WMMA): scores[b,h,s] = (U[b,h,:] . order[b,s,:]) * INV_SCALE
 * Computed as D[16s x 16h] = A[16s x 4k] * B[4k x 16h], K-loop over HIDDEN.
 * A layout: lanes 0-15 -> M rows, vgpr0/1 = K0/K1; lanes 16-31 -> K2/K3.
 * B layout: lanes 0-15 -> N cols, vgpr0/1 = K0/K1; lanes 16-31 -> K2/K3.
 */
__global__ void scores_kernel(const float* __restrict__ order,
                              const float* __restrict__ U,
                              float* __restrict__ scores) {
  int b = blockIdx.y;
  int wave = threadIdx.x >> 5;
  int lane = threadIdx.x & 31;
  int s0 = (blockIdx.x * 8 + wave) * 16;
  const float* ob = order + (size_t)b * SEQ * HIDDEN;
  const float* ub = U + (size_t)b * N_HEADS * HIDDEN;
  int l15 = lane & 15;
  int k2 = (lane >> 4) * 2;
  const float* aptr = ob + (size_t)(s0 + l15) * HIDDEN + k2;
  const float* bptr = ub + (size_t)l15 * HIDDEN + k2;
  v8f c = {};
  #pragma unroll 2
  for (int k0 = 0; k0 < HIDDEN; k0 += 4) {
    v2f a = *(const v2f*)(aptr + k0);
    v2f bb = *(const v2f*)(bptr + k0);
    c = __builtin_amdgcn_wmma_f32_16x16x4_f32(false, a, false, bb,
                                              (short)0, c, false, false);
  }
  /* D: lane n = h = l15; vgpr j -> m = j + 8*(lane>>4) (s within tile) */
  int mbase = (lane >> 4) * 8;
  float* sc = scores + ((size_t)b * N_HEADS + l15) * SEQ + s0 + mbase;
  v4f lo = { c[0] * INV_SCALE, c[1] * INV_SCALE, c[2] * INV_SCALE, c[3] * INV_SCALE };
  v4f hi = { c[4] * INV_SCALE, c[5] * INV_SCALE, c[6] * INV_SCALE, c[7] * INV_SCALE };
  *(v4f*)(sc) = lo;
  *(v4f*)(sc + 4) = hi;
}

/* ---------------- kernel 4: in-place softmax over S per (b,h) */
__global__ void softmax_kernel(float* __restrict__ attn) {
  int b = blockIdx.y, h = blockIdx.x;
  float* p = attn + ((size_t)b * N_HEADS + h) * SEQ;
  __shared__ float red[256];
  int tid = threadIdx.x;
  float x[8];
  float mx = -1e30f;
  #pragma unroll
  for (int j = 0; j < 8; ++j) { x[j] = p[tid + j * 256]; mx = fmaxf(mx, x[j]); }
  red[tid] = mx; __syncthreads();
  for (int off = 128; off > 0; off >>= 1) {
    if (tid < off) red[tid] = fmaxf(red[tid], red[tid + off]);
    __syncthreads();
  }
  mx = red[0]; __syncthreads();
  float s = 0.f;
  #pragma unroll
  for (int j = 0; j < 8; ++j) { x[j] = __expf(x[j] - mx); s += x[j]; }
  red[tid] = s; __syncthreads();
  for (int off = 128; off > 0; off >>= 1) {
    if (tid < off) red[tid] += red[tid + off];
    __syncthreads();
  }
  float inv = 1.0f / red[0];
  #pragma unroll
  for (int j = 0; j < 8; ++j) p[tid + j * 256] = x[j] * inv;
}

/* ---------------- kernel 5: attn_mean[b,s] = mean_h attn[b,h,s] */
__global__ void attnmean_kernel(const float* __restrict__ attn,
                                float* __restrict__ out2) {
  int b = blockIdx.y;
  int s = blockIdx.x * blockDim.x + threadIdx.x;
  float acc = 0.f;
  #pragma unroll
  for (int h = 0; h < N_HEADS; ++h)
    acc += attn[((size_t)b * N_HEADS + h) * SEQ + s];
  out2[(size_t)b * SEQ + s] = acc * (1.0f / N_HEADS);
}

/* ---------------- kernel 6 (WMMA): w2[b,h,n] = sum_s attn[b,h,s] * order[b,s,n]
 * D[16h x 16n] = A[16h x 4s] * B[4s x 16n], K-loop over SEQ.
 */
__global__ void wsum_kernel(const float* __restrict__ order,
                            const float* __restrict__ attn,
                            float* __restrict__ w2) {
  int b = blockIdx.y;
  int wave = threadIdx.x >> 5;
  int lane = threadIdx.x & 31;
  int n0 = (blockIdx.x * 8 + wave) * 16;
  const float* ob = order + (size_t)b * SEQ * HIDDEN;
  const float* ab = attn + (size_t)b * N_HEADS * SEQ;
  int l15 = lane & 15;
  int k2 = (lane >> 4) * 2;
  const float* aptr = ab + (size_t)l15 * SEQ + k2;
  const float* bptr = ob + (size_t)k2 * HIDDEN + n0 + l15;
  v8f c = {};
  #pragma unroll 2
  for (int k0 = 0; k0 < SEQ; k0 += 4) {
    v2f a = *(const v2f*)(aptr + k0);
    v2f bb;
    bb.x = bptr[(size_t)k0 * HIDDEN];           /* row s=k0+k2,   coalesced over 16 lanes */
    bb.y = bptr[(size_t)(k0 + 1) * HIDDEN];     /* row s=k0+k2+1 */
    c = __builtin_amdgcn_wmma_f32_16x16x4_f32(false, a, false, bb,
                                              (short)0, c, false, false);
  }
  int mbase = (lane >> 4) * 8;  /* m = head */
  #pragma unroll
  for (int j = 0; j < 8; ++j)
    w2[((size_t)b * N_HEADS + mbase + j) * HIDDEN + n0 + l15] = c[j];
}

/* ---------------- kernel 7: ctx[b,i] = W_v[i,:] . w2[b, i>>6, :]  (i in [0,1024)) */
__global__ void ctx_kernel(const float* __restrict__ Wv,
                           const float* __restrict__ w2,
                           float* __restrict__ ctx) {
  int b = blockIdx.y;
  int i = blockIdx.x * blockDim.x + threadIdx.x;
  int h = i >> 6;
  const float* w = Wv + (size_t)i * HIDDEN;
  const float* ww = w2 + ((size_t)b * N_HEADS + h) * HIDDEN;
  float acc = 0.f;
  for (int k = 0; k < HIDDEN; k += 4) {
    v4f a = *(const v4f*)(w + k);
    v4f v = *(const v4f*)(ww + k);
    acc += a.x * v.x + a.y * v.y + a.z * v.z + a.w * v.w;
  }
  ctx[(size_t)b * D_MODEL + i] = acc;
}

/* ---------------- kernel 8: out = ctx @ W_o^T + b_o + query; LayerNorm -> y */
__global__ void outln_kernel(const float* __restrict__ Wo,
                             const float* __restrict__ b_o,
                             const float* __restrict__ query,
                             const float* __restrict__ gamma,
                             const float* __restrict__ beta,
                             const float* __restrict__ ctx,
                             float* __restrict__ y) {
  int b = blockIdx.x;
  int tid = threadIdx.x;
  __shared__ float cs[D_MODEL];
  __shared__ float red[256];
  for (int j = tid; j < D_MODEL; j += 256) cs[j] = ctx[(size_t)b * D_MODEL + j];
  __syncthreads();
  float x[8];
  float ssum = 0.f;
  #pragma unroll
  for (int j = 0; j < 8; ++j) {
    int i = tid + j * 256;
    const float* w = Wo + (size_t)i * D_MODEL;
    float acc = 0.f;
    for (int k = 0; k < D_MODEL; k += 4) {
      v4f a = *(const v4f*)(w + k);
      acc += a.x * cs[k] + a.y * cs[k + 1] + a.z * cs[k + 2] + a.w * cs[k + 3];
    }
    x[j] = acc + b_o[i] + query[(size_t)b * HIDDEN + i];
    ssum += x[j];
  }
  red[tid] = ssum; __syncthreads();
  for (int off = 128; off > 0; off >>= 1) {
    if (tid < off) red[tid] += red[tid + off];
    __syncthreads();
  }
  float mu = red[0] * (1.0f / HIDDEN); __syncthreads();
  float vs = 0.f;
  #pragma unroll
  for (int j = 0; j < 8; ++j) { float d = x[j] - mu; vs += d * d; }
  red[tid] = vs; __syncthreads();
  for (int off = 128; off > 0; off >>= 1) {
    if (tid < off) red[tid] += red[tid + off];
    __syncthreads();
  }
  float rstd = rsqrtf(red[0] * (1.0f / HIDDEN) + LN_EPS);
  #pragma unroll
  for (int j = 0; j < 8; ++j) {
    int i = tid + j * 256;
    y[(size_t)b * HIDDEN + i] = (x[j] - mu) * rstd * gamma[i] + beta[i];
  }
}

extern "C" void kernel_launch(void* const* d_in, const int* in_sizes, int n_in,
                              void* d_out, int out_size, void* d_ws, size_t ws_size,
                              hipStream_t stream) {
  const float* query = (const float*)d_in[0];
  const float* order = (const float*)d_in[1];
  const float* Wq    = (const float*)d_in[2];
  const float* Wk    = (const float*)d_in[3];
  const float* Wv    = (const float*)d_in[4];
  const float* Wo    = (const float*)d_in[5];
  const float* bo    = (const float*)d_in[6];
  const float* gamma = (const float*)d_in[7];
  const float* beta  = (const float*)d_in[8];

  float* y         = (float*)d_out;                        /* [32,2048] */
  float* attn_mean = (float*)d_out + BATCH * HIDDEN;       /* [32,2048] */

  float* ws   = (float*)d_ws;
  float* Q    = ws;                                        /* 32*1024   */
  float* U    = Q + BATCH * D_MODEL;                       /* 32*16*2048 */
  float* attn = U + (size_t)BATCH * N_HEADS * HIDDEN;      /* 32*16*2048 */
  float* w2   = attn + (size_t)BATCH * N_HEADS * SEQ;      /* 32*16*2048 */
  float* ctx  = w2 + (size_t)BATCH * N_HEADS * HIDDEN;     /* 32*1024   */

  qproj_kernel  <<<dim3(D_MODEL / 256, BATCH), 256, 0, stream>>>(query, Wq, Q);
  ukey_kernel   <<<dim3(HIDDEN / 256, BATCH), 256, 0, stream>>>(Q, Wk, U);
  scores_kernel <<<dim3(SEQ / 128, BATCH), 256, 0, stream>>>(order, U, attn);
  softmax_kernel<<<dim3(N_HEADS, BATCH), 256, 0, stream>>>(attn);
  attnmean_kernel<<<dim3(SEQ / 256, BATCH), 256, 0, stream>>>(attn, attn_mean);
  wsum_kernel   <<<dim3(HIDDEN / 128, BATCH), 256, 0, stream>>>(order, attn, w2);
  ctx_kernel    <<<dim3(D_MODEL / 256, BATCH), 256, 0, stream>>>(Wv, w2, ctx);
  outln_kernel  <<<BATCH, 256, 0, stream>>>(Wo, bo, query, gamma, beta, ctx, y);
}